// GATNet_3298534884305
// MI455X (gfx1250) — compile-verified
//
#include <hip/hip_runtime.h>
#include <math.h>

#define HID 64
#define NEG_SLOPE 0.2f
#define NEG_INF_ORD 0x007FFFFFu   // order-encoding of -inf

typedef __attribute__((ext_vector_type(2))) float v2f;
typedef __attribute__((ext_vector_type(8))) float v8f;

// ---- order-preserving float<->uint map for atomic float-max via atomicMax(u32) ----
__device__ __forceinline__ unsigned f2ord(float f) {
    unsigned u = __float_as_uint(f);
    return (u & 0x80000000u) ? ~u : (u | 0x80000000u);
}
__device__ __forceinline__ float ord2f(unsigned u) {
    return __uint_as_float((u & 0x80000000u) ? (u & 0x7fffffffu) : ~u);
}

// ------------------------------ fills (run every launch) ------------------------------
__global__ void k_fill_f32(float* __restrict__ p, float v, long n) {
    long i = (long)blockIdx.x * blockDim.x + threadIdx.x;
    if (i < n) p[i] = v;
}
__global__ void k_fill_u32(unsigned* __restrict__ p, unsigned v, long n) {
    long i = (long)blockIdx.x * blockDim.x + threadIdx.x;
    if (i < n) p[i] = v;
}

// ------------------------------ layer-1 GEMM via WMMA f32 16x16x4 ------------------------------
// h1[N,64] = x[N,5] @ W1[5,64], K padded 5 -> 8 (two chained WMMAs).
// 4 waves per block; each wave computes one 16-row tile across 4 column tiles of 16.
__global__ void __launch_bounds__(128) k_h1_wmma(const float* __restrict__ x,
                                                 const float* __restrict__ W1,
                                                 float* __restrict__ h1, int N) {
    const int lane = threadIdx.x & 31;
    const int wave = threadIdx.x >> 5;
    const int half = lane >> 4;           // 0: K={0,1}/{4}; 1: K={2,3}
    const int l    = lane & 15;
    const int rowBase = (blockIdx.x * 4 + wave) * 16;
    if (rowBase >= N) return;             // wave-uniform exit; live waves keep EXEC all-1s

    const int  m  = rowBase + l;          // A-matrix row for this lane
    const int  k0 = half * 2;             // branch-free K base per lane half

    // A tile, f32 16x4 layout: lanes 0-15 hold K={0,1}; lanes 16-31 hold K={2,3}
    v2f a0 = {0.f, 0.f};                  // K chunk {0..3}
    v2f a1 = {0.f, 0.f};                  // K chunk {4..7} (K=5,6,7 zero-padded)
    if (m < N) {
        const float* xr = x + (long)m * 5;
        a0.x = xr[k0];
        a0.y = xr[k0 + 1];
        a1.x = half ? 0.f : xr[4];        // v_cndmask, no EXEC split
    }

    const bool full = (rowBase + 16) <= N;   // wave-uniform (always true for N%16==0)

    #pragma unroll
    for (int t = 0; t < 4; ++t) {
        const int n0 = t * 16 + l;        // B-matrix column for this lane
        v2f b0, b1;
        b0.x = W1[k0 * HID + n0];
        b0.y = W1[(k0 + 1) * HID + n0];
        b1.x = half ? 0.f : W1[4 * HID + n0];
        b1.y = 0.f;

        v8f c = {};
        c = __builtin_amdgcn_wmma_f32_16x16x4_f32(false, a0, false, b0, (short)0, c, false, false);
        c = __builtin_amdgcn_wmma_f32_16x16x4_f32(false, a1, false, b1, (short)0, c, false, false);

        // C/D layout: VGPR r -> M = r + 8*half, N = l
        float* hp = h1 + (long)(rowBase + half * 8) * HID + t * 16 + l;
        if (full) {
            #pragma unroll
            for (int r = 0; r < 8; ++r) hp[(long)r * HID] = c[r];
        } else {
            #pragma unroll
            for (int r = 0; r < 8; ++r)
                if (rowBase + half * 8 + r < N) hp[(long)r * HID] = c[r];
        }
    }
}

// ------------------------------ per-node attention scores, layer 1 ------------------------------
__global__ void k_node_scores1(const float* __restrict__ h1,
                               const float* __restrict__ a_src, const float* __restrict__ a_dst,
                               float* __restrict__ ss, float* __restrict__ sd, int N) {
    int n = blockIdx.x * blockDim.x + threadIdx.x;
    if (n >= N) return;
    const float* hr = h1 + (long)n * HID;
    float as = 0.f, ad = 0.f;
    #pragma unroll 8
    for (int k = 0; k < HID; ++k) { float h = hr[k]; as += h * a_src[k]; ad += h * a_dst[k]; }
    ss[n] = as; sd[n] = ad;
}

// ------------------------------ edge pass 1: segment max (LeakyReLU logits) ------------------------------
__global__ void k_edge_max(const int* __restrict__ src, const int* __restrict__ dst,
                           const float* __restrict__ ss, const float* __restrict__ sd,
                           unsigned* __restrict__ mord, int E, int N) {
    long i = (long)blockIdx.x * blockDim.x + threadIdx.x;
    if (i >= (long)E + N) return;
    int s, d;
    if (i < E) { s = src[i]; d = dst[i]; } else { s = d = (int)(i - E); }  // self loops
    float e = ss[s] + sd[d];
    e = e > 0.f ? e : NEG_SLOPE * e;
    atomicMax(&mord[d], f2ord(e));
}

// ------------------------------ edge pass 2: exp and segment denominator ------------------------------
__global__ void k_edge_expsum(const int* __restrict__ src, const int* __restrict__ dst,
                              const float* __restrict__ ss, const float* __restrict__ sd,
                              const unsigned* __restrict__ mord,
                              float* __restrict__ ex_buf, float* __restrict__ denom, int E, int N) {
    long i = (long)blockIdx.x * blockDim.x + threadIdx.x;
    if (i >= (long)E + N) return;
    int s, d;
    if (i < E) { s = src[i]; d = dst[i]; } else { s = d = (int)(i - E); }
    float e = ss[s] + sd[d];
    e = e > 0.f ? e : NEG_SLOPE * e;
    float ex = expf(e - ord2f(mord[d]));
    ex_buf[i] = ex;
    atomicAdd(&denom[d], ex);
}

// ------------------------------ edge pass 3 (layer1): 64-wide weighted scatter-add ------------------------------
// 16 threads per edge; each handles 4 consecutive features via a float4 gather + 4 f32 atomics.
__global__ void k_edge_aggr64(const int* __restrict__ src, const int* __restrict__ dst,
                              const float* __restrict__ ex_buf, const float* __restrict__ denom,
                              const float* __restrict__ h, float* __restrict__ out, int E, int N) {
    long gid = (long)blockIdx.x * blockDim.x + threadIdx.x;
    long tot = ((long)E + N) * 16;
    if (gid >= tot) return;
    long i = gid >> 4;
    int  g = (int)(gid & 15);
    int s, d;
    if (i < E) { s = src[i]; d = dst[i]; } else { s = d = (int)(i - E); }
    float coef = ex_buf[i] / denom[d];
    const float4 hv = *(const float4*)(h + (long)s * HID + g * 4);
    float* o = out + (long)d * HID + g * 4;
    atomicAdd(o + 0, coef * hv.x);
    atomicAdd(o + 1, coef * hv.y);
    atomicAdd(o + 2, coef * hv.z);
    atomicAdd(o + 3, coef * hv.w);
}

// ------------------------------ bias + ReLU between layers ------------------------------
__global__ void k_relu_bias(float* __restrict__ o, const float* __restrict__ b, long total) {
    long i = (long)blockIdx.x * blockDim.x + threadIdx.x;
    if (i >= total) return;
    float v = o[i] + b[i & (HID - 1)];
    o[i] = v > 0.f ? v : 0.f;
}

// ------------------------------ layer-2 projection + scores + d_out init ------------------------------
__global__ void k_node2(const float* __restrict__ h, const float* __restrict__ W2,
                        const float* __restrict__ a_src2, const float* __restrict__ a_dst2,
                        const float* __restrict__ b2,
                        float* __restrict__ h2, float* __restrict__ ss, float* __restrict__ sd,
                        float* __restrict__ outp, int N) {
    int n = blockIdx.x * blockDim.x + threadIdx.x;
    if (n >= N) return;
    const float* hr = h + (long)n * HID;
    float acc = 0.f;
    #pragma unroll 8
    for (int k = 0; k < HID; ++k) acc += hr[k] * W2[k];
    h2[n] = acc;
    ss[n] = acc * a_src2[0];
    sd[n] = acc * a_dst2[0];
    outp[n] = b2[0];                 // init output with bias (d_out is poisoned)
}

// ------------------------------ edge pass 3 (layer2): scalar weighted scatter-add ------------------------------
__global__ void k_edge_aggr1(const int* __restrict__ src, const int* __restrict__ dst,
                             const float* __restrict__ ex_buf, const float* __restrict__ denom,
                             const float* __restrict__ h2, float* __restrict__ out, int E, int N) {
    long i = (long)blockIdx.x * blockDim.x + threadIdx.x;
    if (i >= (long)E + N) return;
    int s, d;
    if (i < E) { s = src[i]; d = dst[i]; } else { s = d = (int)(i - E); }
    float coef = ex_buf[i] / denom[d];
    atomicAdd(&out[d], coef * h2[s]);
}

static inline unsigned gridFor(long total, int b) { return (unsigned)((total + b - 1) / b); }

extern "C" void kernel_launch(void* const* d_in, const int* in_sizes, int n_in,
                              void* d_out, int out_size, void* d_ws, size_t ws_size,
                              hipStream_t stream) {
    const float* x      = (const float*)d_in[0];
    const int*   ei     = (const int*)d_in[1];   // edge_index [2,E]
    const float* W1     = (const float*)d_in[2];
    const float* a_src1 = (const float*)d_in[3];
    const float* a_dst1 = (const float*)d_in[4];
    const float* b1     = (const float*)d_in[5];
    const float* W2     = (const float*)d_in[6];
    const float* a_src2 = (const float*)d_in[7];
    const float* a_dst2 = (const float*)d_in[8];
    const float* b2     = (const float*)d_in[9];

    const int N = in_sizes[0] / 5;
    const int E = in_sizes[1] / 2;
    const int* src = ei;
    const int* dst = ei + E;
    float* outp = (float*)d_out;

    // ---- workspace carve (floats) ----
    float* ws   = (float*)d_ws;
    float* h1   = ws; ws += (long)N * HID;
    float* out1 = ws; ws += (long)N * HID;
    float* s1s  = ws; ws += N;
    float* s1d  = ws; ws += N;
    float* den1 = ws; ws += N;
    unsigned* m1 = (unsigned*)ws; ws += N;
    float* h2   = ws; ws += N;
    float* s2s  = ws; ws += N;
    float* s2d  = ws; ws += N;
    float* den2 = ws; ws += N;
    unsigned* m2 = (unsigned*)ws; ws += N;
    float* ex1  = ws; ws += (long)E + N;
    float* ex2  = ws; ws += (long)E + N;

    const int  B  = 256;
    const long EN = (long)E + N;

    // ---------------- layer 1 ----------------
    k_h1_wmma<<<dim3((N + 63) / 64), dim3(128), 0, stream>>>(x, W1, h1, N);
    k_fill_f32<<<gridFor((long)N * HID, B), B, 0, stream>>>(out1, 0.f, (long)N * HID);
    k_fill_f32<<<gridFor(N, B), B, 0, stream>>>(den1, 0.f, N);
    k_fill_u32<<<gridFor(N, B), B, 0, stream>>>(m1, NEG_INF_ORD, N);
    k_node_scores1<<<gridFor(N, B), B, 0, stream>>>(h1, a_src1, a_dst1, s1s, s1d, N);
    k_edge_max<<<gridFor(EN, B), B, 0, stream>>>(src, dst, s1s, s1d, m1, E, N);
    k_edge_expsum<<<gridFor(EN, B), B, 0, stream>>>(src, dst, s1s, s1d, m1, ex1, den1, E, N);
    k_edge_aggr64<<<gridFor(EN * 16, B), B, 0, stream>>>(src, dst, ex1, den1, h1, out1, E, N);
    k_relu_bias<<<gridFor((long)N * HID, B), B, 0, stream>>>(out1, b1, (long)N * HID);

    // ---------------- layer 2 ----------------
    k_fill_f32<<<gridFor(N, B), B, 0, stream>>>(den2, 0.f, N);
    k_fill_u32<<<gridFor(N, B), B, 0, stream>>>(m2, NEG_INF_ORD, N);
    k_node2<<<gridFor(N, B), B, 0, stream>>>(out1, W2, a_src2, a_dst2, b2, h2, s2s, s2d, outp, N);
    k_edge_max<<<gridFor(EN, B), B, 0, stream>>>(src, dst, s2s, s2d, m2, E, N);
    k_edge_expsum<<<gridFor(EN, B), B, 0, stream>>>(src, dst, s2s, s2d, m2, ex2, den2, E, N);
    k_edge_aggr1<<<gridFor(EN, B), B, 0, stream>>>(src, dst, ex2, den2, h2, outp, E, N);
}